// SelfAttention_16535624089950
// MI455X (gfx1250) — compile-verified
//
#include <hip/hip_runtime.h>
#include <stdint.h>

typedef __attribute__((ext_vector_type(2))) float v2f;
typedef __attribute__((ext_vector_type(4))) float v4f;
typedef __attribute__((ext_vector_type(8))) float v8f;

// Problem constants (match reference)
#define NB    32      // batch
#define DIM   4096
#define NHQ   32
#define NHKV  8
#define HD    128
#define MAXS  2048
#define NQCOL 4096    // HQ*HD
#define NKCOL 1024    // HKV*HD
#define KTP   132     // padded LDS row pitch (floats) -> conflict-free ds_load_b64
#define SCALE 0.08838834764831845f  // 1/sqrt(128)

static __device__ __forceinline__ v8f wmma_f32_16x16x4(v2f a, v2f b, v8f c) {
  // 8 args: (neg_a, A, neg_b, B, c_mod, C, reuse_a, reuse_b)
  return __builtin_amdgcn_wmma_f32_16x16x4_f32(false, a, false, b, (short)0, c,
                                               false, false);
}

// Byte offset of a __shared__ object inside the workgroup LDS allocation
// (VDST operand of global_load_async_to_lds_* is an LDS byte address).
static __device__ __forceinline__ uint32_t lds_off(const void* p) {
  return (uint32_t)(uintptr_t)(__attribute__((address_space(3))) const void*)p;
}

// Async global->LDS copy, 16B per lane (CDNA5 GLOBAL_LOAD_ASYNC_TO_LDS_B128,
// tracked by ASYNCcnt).
static __device__ __forceinline__ void async_copy_b128(uint32_t lds_byte_addr,
                                                       const float* gaddr) {
  asm volatile("global_load_async_to_lds_b128 %0, %1, off"
               :: "v"(lds_byte_addr), "v"(gaddr)
               : "memory");
}

static __device__ __forceinline__ void wait_async0() {
  asm volatile("s_wait_asynccnt 0x0" ::: "memory");
}
static __device__ __forceinline__ void wait_ds0() {
  asm volatile("s_wait_dscnt 0x0" ::: "memory");
}

// Accumulate two 16x16 f32 tiles (A rows 0-15 and 16-31) over kcount K values.
// Bcol points at column (n-base + lane&15) of the weight matrix (row-major, ld=ldb).
// Pointer-bump form so the unroller emits immediate-offset loads (no per-iter
// 64-bit multiplies).
static __device__ __forceinline__ void wmma_kloop(const float* __restrict__ A, int lda,
                                                  const float* __restrict__ Bcol, int ldb,
                                                  int kbeg, int kcount, int lane,
                                                  v8f& acc0, v8f& acc1) {
  const int m = lane & 15;
  const int h = lane >> 4;
  const float* a0p = A + (size_t)m * lda + kbeg + 2 * h;
  const float* a1p = A + (size_t)(m + 16) * lda + kbeg + 2 * h;
  const float* bp  = Bcol + (size_t)(kbeg + 2 * h) * ldb;
  const size_t bstep = (size_t)4 * ldb;
#pragma unroll 8
  for (int i = 0; i < kcount / 4; ++i) {
    v2f a0 = *(const v2f*)a0p;                 // A[m][kb], A[m][kb+1]
    v2f a1 = *(const v2f*)a1p;
    v2f bv;
    bv.x = bp[0];                              // B[kb][n]
    bv.y = bp[ldb];                            // B[kb+1][n]
    acc0 = wmma_f32_16x16x4(a0, bv, acc0);
    acc1 = wmma_f32_16x16x4(a1, bv, acc1);
    a0p += 4;
    a1p += 4;
    bp  += bstep;
  }
}

// ---------------------------------------------------------------------------
// Kernel 1: fused QKV projection + RoPE.  C(32 x 6144) = x @ [wq | wk | wv].
// 384 blocks (one 16-col tile each), 4 waves split K=4096, LDS reduce.
// q -> q_ws (rope), k -> k_ws (rope), v -> v_ws.  Input caches are NOT written.
// ---------------------------------------------------------------------------
__global__ __launch_bounds__(128) void qkv_rope_wmma(
    const float* __restrict__ x, const float* __restrict__ wq,
    const float* __restrict__ wk, const float* __restrict__ wv,
    const float* __restrict__ rot_cos, const float* __restrict__ rot_sin,
    float* __restrict__ qout, float* __restrict__ kout, float* __restrict__ vout) {
  __shared__ float red[4][32][16];
  const int tid  = threadIdx.x;
  const int lane = tid & 31;
  const int wave = tid >> 5;
  const int col0 = blockIdx.x * 16;

  const float* w; int ldb, mode, lcol0;       // mode: 0=Q(rope) 1=K(rope) 2=V
  if (col0 < NQCOL)                { w = wq; ldb = NQCOL; mode = 0; lcol0 = col0; }
  else if (col0 < NQCOL + NKCOL)   { w = wk; ldb = NKCOL; mode = 1; lcol0 = col0 - NQCOL; }
  else                             { w = wv; ldb = NKCOL; mode = 2; lcol0 = col0 - NQCOL - NKCOL; }

  v8f acc0 = {}, acc1 = {};
  const float* Bcol = w + lcol0 + (lane & 15);
  wmma_kloop(x, DIM, Bcol, ldb, wave * 1024, 1024, lane, acc0, acc1);

#pragma unroll
  for (int j = 0; j < 8; ++j) {
    red[wave][lane][j]     = acc0[j];
    red[wave][lane][8 + j] = acc1[j];
  }
  __syncthreads();

  const int rl = tid & 31;
  const int sb = (tid >> 5) * 4;
  float v4[4];
#pragma unroll
  for (int q = 0; q < 4; ++q) {
    const int s = sb + q;
    v4[q] = red[0][rl][s] + red[1][rl][s] + red[2][rl][s] + red[3][rl][s];
  }
  __syncthreads();
#pragma unroll
  for (int q = 0; q < 4; ++q) red[0][rl][sb + q] = v4[q];
  __syncthreads();

#pragma unroll
  for (int q = 0; q < 4; ++q) {
    const int s = sb + q;
    const int n = rl & 15;
    const int m = (s & 7) + 8 * (rl >> 4) + 16 * (s >> 3);   // batch row 0..31
    const int c = lcol0 + n;                                  // column in target
    float v = red[0][rl][s];
    float outv = v;
    if (mode != 2) {                                          // RoPE on q/k
      const float partner = red[0][rl ^ 1][s];
      const int d = c & (HD - 1);
      const float cs = rot_cos[d >> 1];
      const float sn = rot_sin[d >> 1];
      outv = (d & 1) ? (partner * sn + v * cs)                // imag at odd
                     : (v * cs - partner * sn);               // real at even
    }
    if (mode == 0)      qout[m * NQCOL + c] = outv;
    else if (mode == 1) kout[m * NKCOL + c] = outv;
    else                vout[m * NKCOL + c] = outv;
  }
}

// ---------------------------------------------------------------------------
// Kernel 2: decode attention, one block per (batch, kv-head).  4 waves split
// the key stream with per-wave online softmax.  Each wave stages its 16x128
// f32 K-tile into a private LDS buffer with global_load_async_to_lds_b128
// (one fully-coalesced 512B row per instruction), then feeds WMMA from
// registers while the next tile's async loads are in flight.
// Key s==start_pos is read from the fresh k/v workspace (cache not mutated).
// ---------------------------------------------------------------------------
__global__ __launch_bounds__(128) void attn_decode_wmma(
    const float* __restrict__ qin, const float* __restrict__ knew,
    const float* __restrict__ vnew, const float* __restrict__ ckey,
    const float* __restrict__ cval, const int* __restrict__ sp_ptr,
    float* __restrict__ attout) {
  __shared__ float qs[16][KTP];          // q^T operand, heads 4..15 zeroed
  __shared__ float kt[4][16 * KTP];      // per-wave K-tile staging buffers
  __shared__ float accL[4][32][16];
  __shared__ float mxL[4][4];
  __shared__ float smL[4][4];

  const int tid  = threadIdx.x;
  const int lane = tid & 31;
  const int wave = tid >> 5;
  const int bb   = blockIdx.x >> 3;
  const int g    = blockIdx.x & 7;
  const int sp   = sp_ptr[0];
  const int S    = sp + 1;

  // Stage q (4 heads x 128, pre-scaled by 1/sqrt(HD)); zero heads 4..15.
  for (int i = tid; i < 16 * HD; i += 128) {
    const int r = i >> 7, d = i & (HD - 1);
    float val = 0.f;
    if (r < 4) val = qin[bb * NQCOL + (g * 4 + r) * HD + d] * SCALE;
    qs[r][d] = val;
  }
  __syncthreads();

  const int n = lane & 15;
  const int h = lane >> 4;
  float mrun[4], srun[4], acc[16];
#pragma unroll
  for (int r = 0; r < 4; ++r) { mrun[r] = -3.0e38f; srun[r] = 0.f; }
#pragma unroll
  for (int j = 0; j < 16; ++j) acc[j] = 0.f;

  const int ntiles = (S + 15) >> 4;
  const float* kbn = knew + (size_t)(bb * NHKV + g) * HD;
  const float* vbn = vnew + (size_t)(bb * NHKV + g) * HD;
  const uint32_t ktbase = lds_off(&kt[wave][0]);

  // Stage one 16-row K tile into this wave's LDS buffer (16 async b128 ops,
  // each copying one contiguous 512B key row).
  auto stage_ktile = [&](int s0) {
#pragma unroll
    for (int rr = 0; rr < 16; ++rr) {
      const int s = s0 + rr;
      const float* rowp = (s == sp)
          ? kbn : ckey + ((size_t)(bb * MAXS + s) * NHKV + g) * HD;
      async_copy_b128(ktbase + (uint32_t)(rr * KTP * 4 + lane * 16),
                      rowp + lane * 4);
    }
  };

  if (wave < ntiles) stage_ktile(wave << 4);

  for (int tile = wave; tile < ntiles; tile += 4) {
    const int s0 = tile << 4;

    wait_async0();                                  // this wave's tile is in LDS
    v2f areg[32];
#pragma unroll
    for (int kc = 0; kc < HD; kc += 4)              // conflict-free ds_load_b64
      areg[kc >> 2] = *(const v2f*)(&kt[wave][n * KTP + kc + 2 * h]);
    wait_ds0();                                     // fragments landed in VGPRs
    if (tile + 4 < ntiles) stage_ktile((tile + 4) << 4);  // overlap HBM fetch

    v8f sc = {};
#pragma unroll
    for (int kc = 0; kc < HD; kc += 4) {
      v2f bq = *(const v2f*)(&qs[n][kc + 2 * h]);   // q^T : B 4x16 chunk
      sc = wmma_f32_16x16x4(areg[kc >> 2], bq, sc);
    }
    // mask keys beyond S
#pragma unroll
    for (int j = 0; j < 8; ++j) {
      if (s0 + j + 8 * h >= S) sc[j] = -3.0e38f;
    }
    // per-head tile max (head r lives in lanes r and r+16)
    float vmx = sc[0];
#pragma unroll
    for (int j = 1; j < 8; ++j) vmx = fmaxf(vmx, sc[j]);
    vmx = fmaxf(vmx, __shfl_xor(vmx, 16, 32));
    float nm[4], fsc[4];
#pragma unroll
    for (int r = 0; r < 4; ++r) {
      const float tm   = __shfl(vmx, r, 32);
      const float newm = fmaxf(mrun[r], tm);
      fsc[r]  = __expf(mrun[r] - newm);
      mrun[r] = newm;
      nm[r]   = newm;
    }
#pragma unroll
    for (int j = 0; j < 16; ++j) acc[j] *= fsc[j >> 2];
#pragma unroll
    for (int r = 0; r < 4; ++r) srun[r] *= fsc[r];

    float nmx = nm[0];
    nmx = (n == 1) ? nm[1] : nmx;
    nmx = (n == 2) ? nm[2] : nmx;
    nmx = (n == 3) ? nm[3] : nmx;
    float p[8];
    float rsum = 0.f;
#pragma unroll
    for (int j = 0; j < 8; ++j) {
      const float pv = (n < 4) ? __expf(sc[j] - nmx) : 0.f;
      p[j] = pv;
      rsum += pv;
    }
    rsum += __shfl_xor(rsum, 16, 32);
#pragma unroll
    for (int r = 0; r < 4; ++r) srun[r] += __shfl(rsum, r, 32);

    // P * V : each lane owns dims [lane*4, lane*4+4) for all 4 heads.
#pragma unroll
    for (int mk = 0; mk < 16; ++mk) {
      const int s = s0 + mk;
      const float* vrow = (s == sp)
          ? vbn : cval + ((size_t)(bb * MAXS + s) * NHKV + g) * HD;
      const v4f vv = *(const v4f*)(vrow + lane * 4);
      const int src = (mk >> 3) << 4;
#pragma unroll
      for (int r = 0; r < 4; ++r) {
        const float pr = __shfl(p[mk & 7], src + r, 32);   // p==0 if masked
        acc[r * 4 + 0] += pr * vv[0];
        acc[r * 4 + 1] += pr * vv[1];
        acc[r * 4 + 2] += pr * vv[2];
        acc[r * 4 + 3] += pr * vv[3];
      }
    }
  }

  // cross-wave flash combine
#pragma unroll
  for (int j = 0; j < 16; ++j) accL[wave][lane][j] = acc[j];
  if (lane == 0) {
#pragma unroll
    for (int r = 0; r < 4; ++r) { mxL[wave][r] = mrun[r]; smL[wave][r] = srun[r]; }
  }
  __syncthreads();
  {
    const int rl = tid & 31;
    const int r  = tid >> 5;   // head within group
    const float M = fmaxf(fmaxf(mxL[0][r], mxL[1][r]), fmaxf(mxL[2][r], mxL[3][r]));
    float Sn = 0.f, o0 = 0.f, o1 = 0.f, o2 = 0.f, o3 = 0.f;
#pragma unroll
    for (int w = 0; w < 4; ++w) {
      const float f = __expf(mxL[w][r] - M);
      Sn += smL[w][r] * f;
      o0 += accL[w][rl][r * 4 + 0] * f;
      o1 += accL[w][rl][r * 4 + 1] * f;
      o2 += accL[w][rl][r * 4 + 2] * f;
      o3 += accL[w][rl][r * 4 + 3] * f;
    }
    const float inv = 1.0f / Sn;
    float* dst = attout + (size_t)bb * NQCOL + (g * 4 + r) * HD + rl * 4;
    dst[0] = o0 * inv; dst[1] = o1 * inv; dst[2] = o2 * inv; dst[3] = o3 * inv;
  }
}

// ---------------------------------------------------------------------------
// Kernel 3: output projection.  out(32 x 4096) = att @ wo.
// ---------------------------------------------------------------------------
__global__ __launch_bounds__(128) void out_proj_wmma(
    const float* __restrict__ att, const float* __restrict__ wo,
    float* __restrict__ out) {
  __shared__ float red[4][32][16];
  const int tid  = threadIdx.x;
  const int lane = tid & 31;
  const int wave = tid >> 5;
  const int col0 = blockIdx.x * 16;

  v8f acc0 = {}, acc1 = {};
  const float* Bcol = wo + col0 + (lane & 15);
  wmma_kloop(att, DIM, Bcol, DIM, wave * 1024, 1024, lane, acc0, acc1);

#pragma unroll
  for (int j = 0; j < 8; ++j) {
    red[wave][lane][j]     = acc0[j];
    red[wave][lane][8 + j] = acc1[j];
  }
  __syncthreads();

  const int rl = tid & 31;
  const int sb = (tid >> 5) * 4;
#pragma unroll
  for (int q = 0; q < 4; ++q) {
    const int s = sb + q;
    const float v = red[0][rl][s] + red[1][rl][s] + red[2][rl][s] + red[3][rl][s];
    const int n = rl & 15;
    const int m = (s & 7) + 8 * (rl >> 4) + 16 * (s >> 3);
    out[(size_t)m * DIM + col0 + n] = v;
  }
}

// ---------------------------------------------------------------------------
extern "C" void kernel_launch(void* const* d_in, const int* in_sizes, int n_in,
                              void* d_out, int out_size, void* d_ws, size_t ws_size,
                              hipStream_t stream) {
  (void)in_sizes; (void)n_in; (void)out_size; (void)ws_size;
  const float* x   = (const float*)d_in[0];
  const float* wq  = (const float*)d_in[1];
  const float* wk  = (const float*)d_in[2];
  const float* wv  = (const float*)d_in[3];
  const float* wo  = (const float*)d_in[4];
  const float* ck  = (const float*)d_in[5];
  const float* cv  = (const float*)d_in[6];
  const float* rc  = (const float*)d_in[7];
  const float* rs  = (const float*)d_in[8];
  const int*   sp  = (const int*)d_in[9];

  float* ws   = (float*)d_ws;
  float* q_ws = ws;                 // 32*4096
  float* k_ws = ws + 131072;        // 32*1024
  float* v_ws = ws + 163840;        // 32*1024
  float* a_ws = ws + 196608;        // 32*4096

  qkv_rope_wmma<<<dim3((NQCOL + 2 * NKCOL) / 16), dim3(128), 0, stream>>>(
      x, wq, wk, wv, rc, rs, q_ws, k_ws, v_ws);
  attn_decode_wmma<<<dim3(NB * NHKV), dim3(128), 0, stream>>>(
      q_ws, k_ws, v_ws, ck, cv, sp, a_ws);
  out_proj_wmma<<<dim3(DIM / 16), dim3(128), 0, stream>>>(a_ws, wo, (float*)d_out);
}